// EvolveGCNO_72541997629445
// MI455X (gfx1250) — compile-verified
//
#include <hip/hip_runtime.h>
#include <cstdint>
#include <cstddef>

#define CCH 128         // channels C
#define MT 10           // m-tiles per block in the WMMA GEMM (6250 % 10 == 0)
#define APAD 132        // padded LDS row stride (floats) -> conflict-free ds_load_b64
#define SLABF (16 * APAD)

typedef float v2f __attribute__((ext_vector_type(2)));
typedef float v8f __attribute__((ext_vector_type(8)));

#if defined(__has_builtin)
#if __has_builtin(__builtin_amdgcn_tensor_load_to_lds) && \
    __has_builtin(__builtin_amdgcn_s_wait_tensorcnt)
#define USE_TDM 1
#endif
#endif

#ifdef USE_TDM
typedef unsigned int u32x4 __attribute__((ext_vector_type(4)));
typedef int i32x8 __attribute__((ext_vector_type(8)));
typedef int i32x4 __attribute__((ext_vector_type(4)));

// Issue one TDM load of a 16 x 128 f32 tile (row-major, row stride 128 floats)
// into LDS at lds_byte_off, inserting 4 DWORDs of padding after every 128
// DWORDs so LDS rows are APAD=132 floats apart. D# layout per
// CDNA5 ISA. Executed by one wave; EXEC ignored.
// This toolchain's builtin is the 6-arg clang-23 form:
//   (u32x4 g0, i32x8 g1, i32x4 g2, i32x4 g3, i32x8 extra, i32 cpol)
__device__ inline void tdm_load_slab(const float* gsrc, unsigned lds_byte_off,
                                     unsigned rows_avail) {
  const unsigned long long ga = (unsigned long long)(uintptr_t)gsrc;
  u32x4 g0;
  g0[0] = 1u;                                  // count=1: valid user descriptor
  g0[1] = lds_byte_off;                        // lds_addr (bytes)
  g0[2] = (unsigned)ga;                        // global_addr[31:0]
  g0[3] = (unsigned)((ga >> 32) & 0x01FFFFFFu) // global_addr[56:32]
          | (2u << 30);                        // type = 2 ("image")
  i32x8 g1;
  g1[0] = (2 << 16)                            // data_size = 4 bytes
        | (1 << 20)                            // pad_enable
        | (6 << 22)                            // pad_interval: 128 DWORDs
        | (3 << 25);                           // pad_amount: 4 DWORDs
  g1[1] = (int)(((unsigned)CCH & 0xFFFFu) << 16);    // tensor_dim0[15:0]
  g1[2] = (int)((rows_avail & 0xFFFFu) << 16);       // dim0[31:16]=0 | dim1 lo16
  g1[3] = (int)(((rows_avail >> 16) & 0xFFFFu)       // tensor_dim1 hi16
        | ((unsigned)CCH << 16));                    // tile_dim0 = 128
  g1[4] = 16;                                  // tile_dim1 = 16 rows, tile_dim2 = 0
  g1[5] = CCH;                                 // tensor_dim0_stride = 128 elems
  g1[6] = 0;
  g1[7] = 0;
  i32x4 gz4; gz4[0] = 0; gz4[1] = 0; gz4[2] = 0; gz4[3] = 0;  // 2D: unused
  i32x8 gz8;
  gz8[0] = 0; gz8[1] = 0; gz8[2] = 0; gz8[3] = 0;
  gz8[4] = 0; gz8[5] = 0; gz8[6] = 0; gz8[7] = 0;
  __builtin_amdgcn_tensor_load_to_lds(g0, g1, gz4, gz4, gz8, 0);
}
#endif  // USE_TDM

// ---------------------------------------------------------------------------
__global__ void zero_deg_kernel(unsigned* __restrict__ deg, int n) {
  int i = blockIdx.x * blockDim.x + threadIdx.x;
  if (i < n) deg[i] = 0u;
}

// deg[dst] += 1 per edge (native global_atomic_add_u32).
__global__ void degree_kernel(const int* __restrict__ dst,
                              unsigned* __restrict__ deg, long long e) {
  long long i = (long long)blockIdx.x * blockDim.x + threadIdx.x;
  if (i < e) atomicAdd(&deg[dst[i]], 1u);
}

// dinv[i] = rsqrt(deg[i] + 1)  (+1 = self loop -> always > 0)
__global__ void dinv_kernel(const unsigned* __restrict__ deg,
                            float* __restrict__ dinv, int n) {
  int i = blockIdx.x * blockDim.x + threadIdx.x;
  if (i < n) dinv[i] = rsqrtf((float)deg[i] + 1.0f);
}

// ---------------------------------------------------------------------------
// LSTM single step (h0 = c0 = 0 -> f-gate dead, w_hh irrelevant):
//   w_new = sigmoid(o) * tanh( sigmoid(i) * tanh(g) )
__global__ void lstm_kernel(const float* __restrict__ weight,
                            const float* __restrict__ w_ih,
                            const float* __restrict__ b_ih,
                            const float* __restrict__ b_hh,
                            float* __restrict__ w_new) {
  __shared__ float wrow[CCH];
  const int r = blockIdx.x;
  const int c = threadIdx.x;
  wrow[c] = weight[r * CCH + c];
  __syncthreads();

  const float* wi = w_ih + (size_t)(c) * CCH;
  const float* wg = w_ih + (size_t)(2 * CCH + c) * CCH;
  const float* wo = w_ih + (size_t)(3 * CCH + c) * CCH;
  float di = 0.f, dg = 0.f, dq = 0.f;
#pragma unroll 8
  for (int k = 0; k < CCH; ++k) {
    const float w = wrow[k];
    di += w * wi[k];
    dg += w * wg[k];
    dq += w * wo[k];
  }
  di += b_ih[c] + b_hh[c];
  dg += b_ih[2 * CCH + c] + b_hh[2 * CCH + c];
  dq += b_ih[3 * CCH + c] + b_hh[3 * CCH + c];

  const float si = 1.0f / (1.0f + __expf(-di));
  const float so = 1.0f / (1.0f + __expf(-dq));
  w_new[r * CCH + c] = so * tanhf(si * tanhf(dg));
}

// ---------------------------------------------------------------------------
// xw = x @ w_new, fp32 WMMA (V_WMMA_F32_16X16X4_F32).
//  * wave = one 16-wide n-tile; its B fragments (16 cols x K=128) are preloaded
//    once into 64 VGPRs and reused across MT m-tiles.
//  * the 16x128 A slab is staged in LDS (TDM double-buffered when available),
//    padded to 132-float rows so the per-lane ds_load_b64 A reads are
//    bank-conflict-free.
// Fragment layouts per CDNA5 ISA §7.12.2:
//   A 16x4 f32 : lanes 0-15 -> K{0,1}, lanes 16-31 -> K{2,3}
//   B  4x16    : lanes 0-15 -> rows K{0,1}, lanes 16-31 -> rows K{2,3}
//   C/D 16x16  : VGPR v: M = v (lanes 0-15) / v+8 (lanes 16-31), N = lane&15
__global__ void gemm_wmma_kernel(const float* __restrict__ x,
                                 const float* __restrict__ wn,
                                 float* __restrict__ xw, int mtiles) {
  __shared__ float aslab[2][SLABF];
  const int wave = threadIdx.x >> 5;  // n-tile 0..7
  const int lane = threadIdx.x & 31;
  const int lrow = lane & 15;
  const bool hi = lane >= 16;
  const int koff = hi ? 2 : 0;
  const int n0 = wave * 16;

  const int t0 = blockIdx.x * MT;
  const int tcnt = (mtiles - t0 < MT) ? (mtiles - t0) : MT;
  if (tcnt <= 0) return;

  // ---- preload all B fragments for this wave's n-tile (64 VGPRs) ----
  v2f bf[32];
#pragma unroll
  for (int s = 0; s < 32; ++s) {
    const float* bp = wn + (size_t)(4 * s + koff) * CCH + n0 + lrow;
    bf[s].x = bp[0];
    bf[s].y = bp[CCH];
  }

#ifdef USE_TDM
  if (wave == 0)
    tdm_load_slab(x + (size_t)t0 * 16 * CCH,
                  (unsigned)(uintptr_t)(void*)&aslab[0][0],
                  (unsigned)((mtiles - t0) * 16));
#endif

  for (int t = 0; t < tcnt; ++t) {
    const int buf = t & 1;
#ifdef USE_TDM
    if (wave == 0) {
      if (t + 1 < tcnt) {  // prefetch next slab into the other buffer
        tdm_load_slab(x + (size_t)(t0 + t + 1) * 16 * CCH,
                      (unsigned)(uintptr_t)(void*)&aslab[buf ^ 1][0],
                      (unsigned)((mtiles - t0 - t - 1) * 16));
        __builtin_amdgcn_s_wait_tensorcnt(1);  // oldest (current slab) done
      } else {
        __builtin_amdgcn_s_wait_tensorcnt(0);
      }
    }
    __syncthreads();  // slab `buf` visible to all waves
#else
    {  // cooperative fallback: 16x128 floats = 512 float4, 256 threads
      const float* gs = x + (size_t)(t0 + t) * 16 * CCH;
      for (int f = threadIdx.x; f < 512; f += 256) {
        const int r = f >> 5, c4 = f & 31;
        const float4 v = ((const float4*)(gs + (size_t)r * CCH))[c4];
        *(float4*)&aslab[0][r * APAD + c4 * 4] = v;
      }
      __syncthreads();
    }
#endif

#ifdef USE_TDM
    const float* arow = &aslab[buf][lrow * APAD + koff];
#else
    const float* arow = &aslab[0][lrow * APAD + koff];
#endif
    v8f acc0 = {};
    v8f acc1 = {};
#pragma unroll
    for (int s = 0; s < 32; s += 2) {
      v2f a0 = *(const v2f*)(arow + 4 * s);
      acc0 = __builtin_amdgcn_wmma_f32_16x16x4_f32(
          false, a0, false, bf[s], (short)0, acc0, false, false);
      v2f a1 = *(const v2f*)(arow + 4 * s + 4);
      acc1 = __builtin_amdgcn_wmma_f32_16x16x4_f32(
          false, a1, false, bf[s + 1], (short)0, acc1, false, false);
    }
    const v8f acc = acc0 + acc1;

    const int rbase = (t0 + t) * 16 + (hi ? 8 : 0);
#pragma unroll
    for (int v = 0; v < 8; ++v)
      xw[(size_t)(rbase + v) * CCH + n0 + lrow] = acc[v];

    __syncthreads();  // all reads of slab `buf` done before it is refilled
  }
}

// Scalar fallback for a ragged row tail (unused at N=100000).
__global__ void gemm_tail_kernel(const float* __restrict__ x,
                                 const float* __restrict__ wn,
                                 float* __restrict__ xw, int row0) {
  const int r = row0 + blockIdx.x;
  const int c = threadIdx.x;
  float s = 0.f;
#pragma unroll 8
  for (int k = 0; k < CCH; ++k) s += x[(size_t)r * CCH + k] * wn[k * CCH + c];
  xw[(size_t)r * CCH + c] = s;
}

// ---------------------------------------------------------------------------
// Self-loop term initializes out:  out[i,:] = xw[i,:] * dinv[i]^2
__global__ void init_out_kernel(const float* __restrict__ xw,
                                const float* __restrict__ dinv,
                                float* __restrict__ out, long long nvec) {
  long long i = (long long)blockIdx.x * blockDim.x + threadIdx.x;  // float4 idx
  if (i >= nvec) return;
  const int node = (int)(i >> 5);  // 32 float4 per row
  const float d = dinv[node];
  const float s = d * d;
  const float4 v = ((const float4*)xw)[i];
  float4 r;
  r.x = v.x * s; r.y = v.y * s; r.z = v.z * s; r.w = v.w * s;
  ((float4*)out)[i] = r;
}

// ---------------------------------------------------------------------------
// Edge scatter: one wave per edge. Edge metadata is wave-uniform -> route the
// index through readfirstlane so src/dst/dinv become scalar loads; the vector
// pipe does only the b128 gather + 4 global_atomic_add_f32 per lane.
__global__ void scatter_kernel(const int* __restrict__ src,
                               const int* __restrict__ dst,
                               const float* __restrict__ xw,
                               const float* __restrict__ dinv,
                               float* __restrict__ out, long long e) {
  const long long gw =
      ((long long)blockIdx.x * blockDim.x + threadIdx.x) >> 5;  // wave id
  if (gw >= e) return;  // wave-uniform exit
  const int lane = threadIdx.x & 31;
  const unsigned gwu = (unsigned)__builtin_amdgcn_readfirstlane((int)gw);
  const int s = src[gwu];
  const int d = dst[gwu];
  const float norm = dinv[s] * dinv[d];

  const float4 v = ((const float4*)(xw + (size_t)s * CCH))[lane];
  float* op = out + (size_t)d * CCH + lane * 4;
  unsafeAtomicAdd(op + 0, v.x * norm);
  unsafeAtomicAdd(op + 1, v.y * norm);
  unsafeAtomicAdd(op + 2, v.z * norm);
  unsafeAtomicAdd(op + 3, v.w * norm);
}

// ---------------------------------------------------------------------------
static inline size_t align256(size_t v) { return (v + 255) & ~(size_t)255; }

extern "C" void kernel_launch(void* const* d_in, const int* in_sizes, int n_in,
                              void* d_out, int out_size, void* d_ws,
                              size_t ws_size, hipStream_t stream) {
  const float* x      = (const float*)d_in[0];
  const int*   eidx   = (const int*)d_in[1];
  const float* weight = (const float*)d_in[2];
  const float* w_ih   = (const float*)d_in[3];
  // d_in[4] = w_hh: zero forward contribution (h0 = 0), unused.
  const float* b_ih   = (const float*)d_in[5];
  const float* b_hh   = (const float*)d_in[6];
  float* out = (float*)d_out;

  const int       N = in_sizes[0] / CCH;  // 100000
  const long long E = in_sizes[1] / 2;    // 1600000
  const int* src = eidx;
  const int* dst = eidx + E;

  // Workspace carve-up.
  char* ws = (char*)d_ws;
  size_t off = 0;
  float*    w_new = (float*)(ws + off);    off += align256((size_t)CCH * CCH * 4);
  unsigned* deg   = (unsigned*)(ws + off); off += align256((size_t)N * 4);
  float*    dinv  = (float*)(ws + off);    off += align256((size_t)N * 4);
  float*    xw    = (float*)(ws + off);    // N*CCH*4 bytes

  // --- degree / normalization ---
  zero_deg_kernel<<<(N + 255) / 256, 256, 0, stream>>>(deg, N);
  degree_kernel<<<(unsigned)((E + 255) / 256), 256, 0, stream>>>(dst, deg, E);
  dinv_kernel<<<(N + 255) / 256, 256, 0, stream>>>(deg, dinv, N);

  // --- evolve the GCN weight via the LSTM step ---
  lstm_kernel<<<CCH, CCH, 0, stream>>>(weight, w_ih, b_ih, b_hh, w_new);

  // --- xw = x @ w_new  (fp32 WMMA, TDM-staged A slabs) ---
  const int mtiles = N / 16;
  if (mtiles > 0) {
    const int mgroups = (mtiles + MT - 1) / MT;
    gemm_wmma_kernel<<<mgroups, 256, 0, stream>>>(x, w_new, xw, mtiles);
  }
  const int rem = N - mtiles * 16;
  if (rem > 0)
    gemm_tail_kernel<<<rem, CCH, 0, stream>>>(x, w_new, xw, mtiles * 16);

  // --- self-loop init:  out = xw * dinv^2 ---
  const long long nvec = (long long)N * (CCH / 4);
  init_out_kernel<<<(unsigned)((nvec + 255) / 256), 256, 0, stream>>>(
      xw, dinv, out, nvec);

  // --- edge scatter: one wave per edge ---
  const long long sthreads = E * 32;
  scatter_kernel<<<(unsigned)((sthreads + 255) / 256), 256, 0, stream>>>(
      src, dst, xw, dinv, out, E);
}